// _YOLO_4569845203300
// MI455X (gfx1250) — compile-verified
//
#include <hip/hip_runtime.h>

#define N_IMG 8
#define HW    4096
#define NA    9
#define NG    64
#define NC    80
#define BLK   256
#define BLKS_PER_IMG (HW / BLK)       // 16
#define NBLK (N_IMG * BLKS_PER_IMG)   // 128

// ---- CDNA5 async global->LDS staging (guarded; falls back to plain loads) ----
#if defined(__gfx1250__) && \
    __has_builtin(__builtin_amdgcn_global_load_async_to_lds_b128) && \
    __has_builtin(__builtin_amdgcn_global_load_async_to_lds_b32)
#define HAVE_ASYNC_LDS 1
#else
#define HAVE_ASYNC_LDS 0
#endif

#if HAVE_ASYNC_LDS
typedef int v4i_t __attribute__((ext_vector_type(4)));
typedef __attribute__((address_space(1))) v4i_t* gptr_v4i;
typedef __attribute__((address_space(3))) v4i_t* lptr_v4i;
typedef __attribute__((address_space(1))) int*   gptr_i32;
typedef __attribute__((address_space(3))) int*   lptr_i32;
#endif

__device__ __forceinline__ void wait_async_zero() {
#if HAVE_ASYNC_LDS
#if __has_builtin(__builtin_amdgcn_s_wait_asynccnt)
  __builtin_amdgcn_s_wait_asynccnt(0);
#else
  asm volatile("s_wait_asynccnt 0" ::: "memory");
#endif
#endif
}

__global__ __launch_bounds__(BLK) void yolo_loss_stage1(
    const float* __restrict__ pred_boxes,   // [N,HW,A,4] xywh
    const float* __restrict__ pred_o,       // [N,HW,A]
    const float* __restrict__ pred_scores,  // [N,HW,A,C]
    const float* __restrict__ gt_boxes,     // [N,G,4] xyxy
    const int*   __restrict__ gt_labels,    // [N,G]
    float* __restrict__ ws)                 // [NBLK,3] partials
{
  __shared__ float4 s_gt[NG];
  __shared__ float  s_ga[NG];
  __shared__ int    s_lbl[NG];
  __shared__ float  s_red[BLK / 32][3];

  const int tid  = threadIdx.x;
  const int img  = blockIdx.x / BLKS_PER_IMG;
  const int cell = (blockIdx.x % BLKS_PER_IMG) * BLK + tid;

  // ---- stage GT boxes + labels into LDS (async global->LDS path on CDNA5) ----
  if (tid < NG) {
    const float4* gsrc = (const float4*)gt_boxes + (size_t)img * NG + tid;
    const int*    lsrc = gt_labels + (size_t)img * NG + tid;
#if HAVE_ASYNC_LDS
    __builtin_amdgcn_global_load_async_to_lds_b128(
        (gptr_v4i)(void*)gsrc, (lptr_v4i)(void*)&s_gt[tid], 0, 0);
    __builtin_amdgcn_global_load_async_to_lds_b32(
        (gptr_i32)(void*)lsrc, (lptr_i32)(void*)&s_lbl[tid], 0, 0);
#else
    s_gt[tid]  = *gsrc;
    s_lbl[tid] = *lsrc;
#endif
  }

  // prefetch this lane's 9 anchor boxes (global_prefetch_b8)
  const float4* pb = (const float4*)pred_boxes + ((size_t)img * HW + cell) * NA;
  __builtin_prefetch(pb, 0, 0);

  wait_async_zero();
  __syncthreads();
  if (tid < NG) {
    float4 b = s_gt[tid];
    s_ga[tid] = (b.z - b.x) * (b.w - b.y);
  }
  __syncthreads();

  // ---- load 9 anchor boxes, convert xywh -> xyxy, precompute areas ----
  float4 bx[NA];
  float  pa[NA];
#pragma unroll
  for (int a = 0; a < NA; ++a) {
    float4 b = pb[a];
    b.z += b.x;
    b.w += b.y;
    bx[a] = b;
    pa[a] = (b.z - b.x) * (b.w - b.y);
  }

  // ---- pass A: per-anchor max IoU over all GT ----
  float amax[NA];
#pragma unroll
  for (int a = 0; a < NA; ++a) amax[a] = -1.0f;  // IoU >= 0 always
  for (int g = 0; g < NG; ++g) {
    const float4 gb = s_gt[g];
    const float  ga = s_ga[g];
#pragma unroll
    for (int a = 0; a < NA; ++a) {
      float lx = fmaxf(bx[a].x, gb.x);
      float ly = fmaxf(bx[a].y, gb.y);
      float rx = fminf(bx[a].z, gb.z);
      float ry = fminf(bx[a].w, gb.w);
      float iw = fmaxf(rx - lx, 0.0f);
      float ih = fmaxf(ry - ly, 0.0f);
      float inter = iw * ih;
      float iou = inter / (pa[a] + ga - inter);
      amax[a] = fmaxf(amax[a], iou);
    }
  }

  // first-occurrence argmax over anchors (strict > matches jnp.argmax)
  int   asel  = 0;
  float abest = amax[0];
#pragma unroll
  for (int a = 1; a < NA; ++a)
    if (amax[a] > abest) { abest = amax[a]; asel = a; }

  const float m = (abest > 0.0f) ? 1.0f : 0.0f;

  // select anchor box/area without dynamic register indexing (cndmask chain)
  float4 sb = bx[0];
  float  sa = pa[0];
#pragma unroll
  for (int a = 1; a < NA; ++a)
    if (asel == a) { sb = bx[a]; sa = pa[a]; }

  // ---- pass B: best GT for selected anchor (first-occurrence argmax) ----
  float biou = -1.0f;
  int   bg   = 0;
  for (int g = 0; g < NG; ++g) {
    const float4 gb = s_gt[g];
    float lx = fmaxf(sb.x, gb.x);
    float ly = fmaxf(sb.y, gb.y);
    float rx = fminf(sb.z, gb.z);
    float ry = fminf(sb.w, gb.w);
    float iw = fmaxf(rx - lx, 0.0f);
    float ih = fmaxf(ry - ly, 0.0f);
    float inter = iw * ih;
    float iou = inter / (sa + s_ga[g] - inter);
    if (iou > biou) { biou = iou; bg = g; }
  }

  // ---- objectness ----
  const float* po = pred_o + ((size_t)img * HW + cell) * NA;
  float ov[NA];
#pragma unroll
  for (int a = 0; a < NA; ++a) ov[a] = po[a];
  float o_sel = ov[0];
  float o_max = ov[0];
#pragma unroll
  for (int a = 1; a < NA; ++a) {
    if (asel == a) o_sel = ov[a];
    o_max = fmaxf(o_max, ov[a]);
  }
  float od  = o_sel - biou;
  float obj = m * od * od + 0.5f * (1.0f - m) * o_max * o_max;

  // ---- bbox regression ----
  const float4 gbb = s_gt[bg];
  float dx = sb.x - gbb.x;
  float dy = sb.y - gbb.y;
  float dz = sqrtf(sb.z) - sqrtf(gbb.z);
  float dw = sqrtf(sb.w) - sqrtf(gbb.w);
  float bbox = m * (dx * dx + dy * dy + dz * dz + dw * dw);

  // ---- classification CE for selected anchor only (1/9 of score bytes) ----
  const float* sc = pred_scores + (((size_t)img * HW + cell) * NA + asel) * NC;
  const float4* sc4 = (const float4*)sc;
  float smax = -3.402823466e+38f;
#pragma unroll 5
  for (int c = 0; c < NC / 4; ++c) {
    float4 v = sc4[c];
    smax = fmaxf(smax, fmaxf(fmaxf(v.x, v.y), fmaxf(v.z, v.w)));
  }
  float ssum = 0.0f;
#pragma unroll 5
  for (int c = 0; c < NC / 4; ++c) {
    float4 v = sc4[c];
    ssum += expf(v.x - smax) + expf(v.y - smax) + expf(v.z - smax) + expf(v.w - smax);
  }
  float lse = smax + logf(ssum);
  int   lbl = s_lbl[bg];
  float clf = m * (lse - sc[lbl]);

  // ---- deterministic block reduction: wave shuffles then LDS ----
  float v0 = obj, v1 = bbox, v2 = clf;
#pragma unroll
  for (int off = 16; off > 0; off >>= 1) {
    v0 += __shfl_down(v0, off, 32);
    v1 += __shfl_down(v1, off, 32);
    v2 += __shfl_down(v2, off, 32);
  }
  const int wave = tid >> 5;
  if ((tid & 31) == 0) {
    s_red[wave][0] = v0;
    s_red[wave][1] = v1;
    s_red[wave][2] = v2;
  }
  __syncthreads();
  if (tid == 0) {
    float a0 = 0.0f, a1 = 0.0f, a2 = 0.0f;
#pragma unroll
    for (int w = 0; w < BLK / 32; ++w) {
      a0 += s_red[w][0];
      a1 += s_red[w][1];
      a2 += s_red[w][2];
    }
    ws[blockIdx.x * 3 + 0] = a0;
    ws[blockIdx.x * 3 + 1] = a1;
    ws[blockIdx.x * 3 + 2] = a2;
  }
}

__global__ __launch_bounds__(32) void yolo_loss_stage2(
    const float* __restrict__ ws, float* __restrict__ out)
{
  const int lane = threadIdx.x;
  float a0 = 0.0f, a1 = 0.0f, a2 = 0.0f;
  for (int b = lane; b < NBLK; b += 32) {
    a0 += ws[b * 3 + 0];
    a1 += ws[b * 3 + 1];
    a2 += ws[b * 3 + 2];
  }
#pragma unroll
  for (int off = 16; off > 0; off >>= 1) {
    a0 += __shfl_down(a0, off, 32);
    a1 += __shfl_down(a1, off, 32);
    a2 += __shfl_down(a2, off, 32);
  }
  if (lane == 0) {
    out[0] = a0;
    out[1] = a1;
    out[2] = a2;
  }
}

extern "C" void kernel_launch(void* const* d_in, const int* in_sizes, int n_in,
                              void* d_out, int out_size, void* d_ws, size_t ws_size,
                              hipStream_t stream) {
  const float* pred_boxes  = (const float*)d_in[0];
  const float* pred_o      = (const float*)d_in[1];
  const float* pred_scores = (const float*)d_in[2];
  const float* gt_boxes    = (const float*)d_in[3];
  const int*   gt_labels   = (const int*)d_in[4];
  float* ws  = (float*)d_ws;
  float* out = (float*)d_out;

  hipLaunchKernelGGL(yolo_loss_stage1, dim3(NBLK), dim3(BLK), 0, stream,
                     pred_boxes, pred_o, pred_scores, gt_boxes, gt_labels, ws);
  hipLaunchKernelGGL(yolo_loss_stage2, dim3(1), dim3(32), 0, stream, ws, out);
}